// Ernie4_5_MoeTopKRouter_57801669870158
// MI455X (gfx1250) — compile-verified
//
#include <hip/hip_runtime.h>
#include <hip/hip_bf16.h>
#include <math.h>

typedef __attribute__((ext_vector_type(2))) float v2f;
typedef __attribute__((ext_vector_type(8))) float v8f;
typedef __attribute__((ext_vector_type(4))) int   v4i;

#define TOKENS   8192
#define HIDDEN   2560
#define NEXP     64
#define TOPK     6
#define NORM_MIN 1e-12f

#define KC       64                 // K-chunk staged in LDS
#define PAD_A    4                  // row pad (floats) -> conflict-free b64 frags
#define LDA      (KC + PAD_A)       // 68
#define PAD_L    2
#define LDL      (NEXP + PAD_L)     // 66
#define WAVES    4
#define ROWS     (WAVES * 16)       // 64 tokens per workgroup
#define NTHREADS (WAVES * 32)       // 128
#define NCHUNK   (HIDDEN / KC)      // 40
#define A_IT     ((ROWS * (KC/4)) / NTHREADS)   // 8 float4 per thread (hidden)
#define B_IT     ((NEXP * (KC/4)) / NTHREADS)   // 8 float4 per thread (weight)
#define LD_PER_THREAD (A_IT + B_IT)             // 16 async loads / thread / chunk

#if defined(__HIP_DEVICE_COMPILE__) && __has_builtin(__builtin_amdgcn_global_load_async_to_lds_b128)
#define HAVE_ASYNC_LDS 1
#else
#define HAVE_ASYNC_LDS 0
#endif

// One 16-byte global->LDS copy. Async (ASYNCcnt, no VGPR round-trip) when the
// gfx1250 builtin exists; plain copy fallback otherwise.
__device__ __forceinline__ void copy16_to_lds(const float* g, float* l) {
#if HAVE_ASYNC_LDS
    // Builtin takes generic v4i* params: (lds_dst, global_src, imm_offset, cpol)
    __builtin_amdgcn_global_load_async_to_lds_b128(
        (v4i*)l, (v4i*)g, /*imm_offset=*/0, /*cpol=*/0);
#else
    *reinterpret_cast<float4*>(l) = *reinterpret_cast<const float4*>(g);
#endif
}

__device__ __forceinline__ void wait_async_le_0() {
#if HAVE_ASYNC_LDS
#if __has_builtin(__builtin_amdgcn_s_wait_asynccnt)
    __builtin_amdgcn_s_wait_asynccnt(0);
#else
    asm volatile("s_wait_asynccnt 0x0" ::: "memory");
#endif
#endif
}

// Allow the next chunk's LD_PER_THREAD loads to remain in flight; our own
// current-chunk loads (issued earlier, retired in order) are then complete.
__device__ __forceinline__ void wait_async_le_pending() {
#if HAVE_ASYNC_LDS
#if __has_builtin(__builtin_amdgcn_s_wait_asynccnt)
    __builtin_amdgcn_s_wait_asynccnt(LD_PER_THREAD);
#else
    asm volatile("s_wait_asynccnt 0x10" ::: "memory");   // 16
#endif
#endif
}

__device__ __forceinline__ void issue_chunk(const float* __restrict__ hid,
                                            const float* __restrict__ wgt,
                                            float (*bA)[LDA], float (*bB)[LDA],
                                            int row0, int k0, int tid) {
    // Fixed trip counts (no data-dependent bound -> no exec predication).
    #pragma unroll
    for (int it = 0; it < A_IT; ++it) {
        const int i  = tid + it * NTHREADS;   // 0..1023
        const int r  = i >> 4;                // row 0..63
        const int c4 = i & 15;                // float4 column
        copy16_to_lds(&hid[(size_t)(row0 + r) * HIDDEN + k0 + c4 * 4],
                      &bA[r][c4 * 4]);
    }
    #pragma unroll
    for (int it = 0; it < B_IT; ++it) {
        const int i  = tid + it * NTHREADS;
        const int r  = i >> 4;
        const int c4 = i & 15;
        copy16_to_lds(&wgt[(size_t)r * HIDDEN + k0 + c4 * 4],
                      &bB[r][c4 * 4]);
    }
}

__global__ __launch_bounds__(NTHREADS)
void moe_router_kernel(const float* __restrict__ hid,
                       const float* __restrict__ wgt,
                       const float* __restrict__ bias,
                       float* __restrict__ logits_out,
                       int*   __restrict__ sel_out,
                       float* __restrict__ rw_out)
{
    __shared__ float sA[2][ROWS][LDA];   // double-buffered hidden tile
    __shared__ float sB[2][NEXP][LDA];   // double-buffered weight tile
    __shared__ float sL[ROWS][LDL];      // logits tile for the epilogue

    const int tid  = threadIdx.x;
    const int lane = tid & 31;
    const int wave = tid >> 5;
    const int row0 = blockIdx.x * ROWS;     // token base of this WG

    const int halfsel = lane >> 4;          // 0: lanes 0-15, 1: lanes 16-31
    const int l16     = lane & 15;
    const int koff    = halfsel * 2;        // K sub-offset inside 4-wide step
    const int arow    = wave * 16 + l16;    // A row this lane feeds

    v8f acc[4];
    #pragma unroll
    for (int e = 0; e < 4; ++e) acc[e] = (v8f){0.f,0.f,0.f,0.f,0.f,0.f,0.f,0.f};

    // Prologue: start chunk 0 into buffer 0.
    issue_chunk(hid, wgt, sA[0], sB[0], row0, 0, tid);

    for (int c = 0; c < NCHUNK; ++c) {
        const int cur = c & 1;

        if (c + 1 < NCHUNK) {
            // Overlap: start next chunk into the other buffer (freed by the
            // barrier at the end of the previous iteration).
            issue_chunk(hid, wgt, sA[1 - cur], sB[1 - cur],
                        row0, (c + 1) * KC, tid);
            wait_async_le_pending();
        } else {
            wait_async_le_0();
        }
        __syncthreads();   // all waves' current-chunk LDS data visible

        // ---- WMMA f32 16x16x4 over the chunk ----
        #pragma unroll 4
        for (int kk = 0; kk < KC; kk += 4) {
            const v2f a = *reinterpret_cast<const v2f*>(&sA[cur][arow][kk + koff]);
            #pragma unroll
            for (int e = 0; e < 4; ++e) {
                const v2f b = *reinterpret_cast<const v2f*>(
                    &sB[cur][e * 16 + l16][kk + koff]);
                acc[e] = __builtin_amdgcn_wmma_f32_16x16x4_f32(
                    /*neg_a=*/false, a, /*neg_b=*/false, b,
                    /*c_mod=*/(short)0, acc[e],
                    /*reuse_a=*/false, /*reuse_b=*/false);
            }
        }
        __syncthreads();   // frees buf[cur] for re-issue in iteration c+1
    }

    // ---- scatter accumulators: C/D layout -> LDS logits + global logits ----
    // v8f elem r, lane l: token = wave*16 + r + 8*(l>>4), expert = e*16 + (l&15)
    #pragma unroll
    for (int e = 0; e < 4; ++e) {
        const int ecol = e * 16 + l16;
        #pragma unroll
        for (int r = 0; r < 8; ++r) {
            const int trow = wave * 16 + r + 8 * halfsel;
            const float v = acc[e][r];
            sL[trow][ecol] = v;
            logits_out[(size_t)(row0 + trow) * NEXP + ecol] = v;
        }
    }
    __syncthreads();

    // ---- phase 2: one thread per token — softmax + top-6 + normalize ----
    if (tid < ROWS) {
        const int tok = row0 + tid;

        float mx = -INFINITY;
        #pragma unroll 8
        for (int j = 0; j < NEXP; ++j) mx = fmaxf(mx, sL[tid][j]);

        float sum = 0.f;
        #pragma unroll 8
        for (int j = 0; j < NEXP; ++j) {
            const float p = __expf(sL[tid][j] - mx);
            sL[tid][j] = p;           // cache unnormalized prob
            sum += p;
        }
        const float invsum = 1.f / sum;

        unsigned long long chosen = 0ull;
        int   idx[TOPK];
        float gat[TOPK];
        float gsum = 0.f;
        #pragma unroll
        for (int t = 0; t < TOPK; ++t) {
            float best = -INFINITY;
            int   bi   = 0;
            for (int j = 0; j < NEXP; ++j) {
                const float cval = sL[tid][j] * invsum + bias[j];
                const bool taken = (chosen >> j) & 1ull;
                if (!taken && cval > best) { best = cval; bi = j; }
            }
            chosen |= (1ull << bi);
            idx[t] = bi;
            const float g = sL[tid][bi] * invsum;
            gat[t] = g;
            gsum  += g;
        }
        const float denom = fmaxf(gsum, NORM_MIN);
        const float inv_d = 1.f / denom;
        #pragma unroll
        for (int t = 0; t < TOPK; ++t) {
            sel_out[(size_t)tok * TOPK + t] = idx[t];
            rw_out [(size_t)tok * TOPK + t] = gat[t] * inv_d;
        }
    }
}

extern "C" void kernel_launch(void* const* d_in, const int* in_sizes, int n_in,
                              void* d_out, int out_size, void* d_ws, size_t ws_size,
                              hipStream_t stream) {
    const float* hid  = (const float*)d_in[0];   // [8192, 2560]
    const float* wgt  = (const float*)d_in[1];   // [64, 2560]
    const float* bias = (const float*)d_in[2];   // [64]

    float* logits = (float*)d_out;                               // 8192*64 f32
    int*   sel    = (int*)((float*)d_out + (size_t)TOKENS*NEXP); // 8192*6  i32
    float* rw     = (float*)d_out + (size_t)TOKENS*NEXP
                                  + (size_t)TOKENS*TOPK;         // 8192*6  f32

    dim3 grid(TOKENS / ROWS);   // 128 workgroups
    dim3 block(NTHREADS);       // 4 waves (wave32)
    moe_router_kernel<<<grid, block, 0, stream>>>(hid, wgt, bias, logits, sel, rw);
}